// MultigridLayer2_74783970558226
// MI455X (gfx1250) — compile-verified
//
#include <hip/hip_runtime.h>

// ---------------- problem constants (mirror reference) ----------------
#define D0c    32
#define D1c    32
#define Gc     1024          // D0*D1
#define Kc     5             // n_orders
#define Vc     5120          // G*K
#define NEQIV  1056          // G + D1
#define NDER   992           // rows per derivative sub-block
#define RTOT   8992          // NEQIV + 8*NDER
#define DSc    1000.0f
#define ALPHAc 0.1f
#define BTOT   4             // BS*NID
#define NB     64            // Cholesky block size
#define NBLK   (Vc / NB)     // 80
#define MT     128           // SYRK macro-tile (per workgroup)
#define LDP    (NB + 4)      // padded LDS row stride (16B aligned, conflict-free)

typedef __attribute__((ext_vector_type(2))) float v2f;
typedef __attribute__((ext_vector_type(8))) float v8f;

// PAIRS = [(0,1,0),(0,2,1),(1,3,0),(2,4,1)]
__device__ __constant__ int PAIR_S[4] = {0, 0, 1, 2};
__device__ __constant__ int PAIR_D[4] = {1, 2, 3, 4};
__device__ __constant__ int PAIR_C[4] = {0, 1, 0, 1};

// -------- sparse row structure of A (per constraint row) --------
__device__ inline int row_entries(int r, const float* coeffs0,
                                  const float* s0, const float* s1,
                                  int* cols, float* vals, float* pinv) {
    if (r < Gc) {                              // equation row
        #pragma unroll
        for (int k = 0; k < Kc; ++k) { cols[k] = r * Kc + k; vals[k] = coeffs0[r * Kc + k]; }
        *pinv = 1.0f;
        return Kc;
    }
    if (r < NEQIV) {                           // initial-condition row
        int j = r - Gc;
        cols[0] = j * Kc; vals[0] = 1.0f; *pinv = 1.0f;
        return 1;
    }
    int t   = r - NEQIV;
    int blk = t / NDER;
    int i   = t - blk * NDER;
    int p   = blk >> 1;
    int nodeSel = blk & 1;
    int s = PAIR_S[p], d = PAIR_D[p], c = PAIR_C[p];
    int g, gp; float h;
    if (c == 0) { g = i;  gp = i + D1c;  h = s0[i >> 5]; }
    else { int I0 = i / (D1c - 1); int I1 = i - I0 * (D1c - 1);
           g = I0 * D1c + I1; gp = g + 1; h = s1[I1]; }
    int node = nodeSel ? gp : g;
    cols[0] = gp * Kc + s;   vals[0] =  1.0f;
    cols[1] = g  * Kc + s;   vals[1] = -1.0f;
    cols[2] = node * Kc + d; vals[2] = -h;
    *pinv = 1.0f / DSc;
    return 3;
}

// ---------------- assembly kernels ----------------
__global__ void zero_ws_kernel(float* p, size_t n) {
    size_t i = (size_t)blockIdx.x * blockDim.x + threadIdx.x;
    size_t stride = (size_t)gridDim.x * blockDim.x;
    for (; i < n; i += stride) p[i] = 0.0f;
}

__global__ void assemble_AtA_kernel(const float* coeffs0, const float* s0,
                                    const float* s1, float* M) {
    int r = blockIdx.x * blockDim.x + threadIdx.x;
    if (r >= RTOT) return;
    int cols[5]; float vals[5]; float pinv;
    int nc = row_entries(r, coeffs0, s0, s1, cols, vals, &pinv);
    for (int a = 0; a < nc; ++a)
        for (int b = 0; b < nc; ++b)
            atomicAdd(&M[(size_t)cols[a] * Vc + cols[b]], pinv * vals[a] * vals[b]);
}

__global__ void add_alpha_diag_kernel(float* M) {
    int i = blockIdx.x * blockDim.x + threadIdx.x;
    if (i < Vc) M[(size_t)i * Vc + i] += ALPHAc;
}

// b = A^T Pinv A_rhs  (only first NEQIV rows have nonzero rhs; Pinv==1 there)
__global__ void assemble_rhs_kernel(const float* coeffs, const float* rhs,
                                    const float* iv, float* bvec) {
    int idx = blockIdx.x * blockDim.x + threadIdx.x;
    if (idx >= BTOT * NEQIV) return;
    int b = idx / NEQIV, r = idx - b * NEQIV;
    if (r < Gc) {
        float w = rhs[b * Gc + r];
        const float* cf = coeffs + (size_t)b * Gc * Kc + (size_t)r * Kc;
        #pragma unroll
        for (int k = 0; k < Kc; ++k)
            atomicAdd(&bvec[(size_t)b * Vc + r * Kc + k], w * cf[k]);
    } else {
        int j = r - Gc;
        atomicAdd(&bvec[(size_t)b * Vc + j * Kc], iv[b * D1c + j]);
    }
}

// ---------------- blocked Cholesky (lower, in place) ----------------
__global__ void chol_potf2_kernel(float* M, int off) {
    __shared__ float s[NB * NB];
    for (int idx = threadIdx.x; idx < NB * NB; idx += blockDim.x)
        s[idx] = M[(size_t)(off + idx / NB) * Vc + off + (idx % NB)];
    __syncthreads();
    for (int j = 0; j < NB; ++j) {
        if (threadIdx.x == 0) s[j * NB + j] = sqrtf(s[j * NB + j]);
        __syncthreads();
        float dj = s[j * NB + j];
        for (int i = j + 1 + threadIdx.x; i < NB; i += blockDim.x) s[i * NB + j] /= dj;
        __syncthreads();
        int w = NB - j - 1;
        for (int idx = threadIdx.x; idx < w * w; idx += blockDim.x) {
            int ii = j + 1 + idx / w, cc = j + 1 + idx % w;
            if (ii >= cc) s[ii * NB + cc] -= s[ii * NB + j] * s[cc * NB + j];
        }
        __syncthreads();
    }
    for (int idx = threadIdx.x; idx < NB * NB; idx += blockDim.x)
        M[(size_t)(off + idx / NB) * Vc + off + (idx % NB)] = s[idx];
}

// panel rows: X * Lkk^T = A  (row-wise forward substitution; row staged in LDS)
__global__ void __launch_bounds__(128)
chol_trsm_kernel(float* M, int off) {
    __shared__ float Lkk[NB * NB];          // 16 KB
    __shared__ float xr[128][NB + 1];       // 33 KB, +1 pad: conflict-free columns
    for (int idx = threadIdx.x; idx < NB * NB; idx += 128)
        Lkk[idx] = M[(size_t)(off + idx / NB) * Vc + off + (idx % NB)];
    int row = off + NB + blockIdx.x * 128 + threadIdx.x;
    bool valid = row < Vc;
    int rc = valid ? row : (Vc - 1);
    const float* a = &M[(size_t)rc * Vc + off];
    float* x = xr[threadIdx.x];
    #pragma unroll 4
    for (int t = 0; t < NB; ++t) x[t] = a[t];
    __syncthreads();
    for (int j = 0; j < NB; ++j) {
        float acc = x[j];
        for (int t = 0; t < j; ++t) acc -= x[t] * Lkk[j * NB + t];
        x[j] = acc / Lkk[j * NB + j];
    }
    if (valid) {
        float* o = &M[(size_t)row * Vc + off];
        #pragma unroll 4
        for (int t = 0; t < NB; ++t) o[t] = x[t];
    }
}

// trailing update: 128x128 macro-tile per workgroup.
// Panels staged into LDS via async DMA (GLOBAL_LOAD_ASYNC_TO_LDS_B128, ASYNCcnt).
// 8 waves in a 4x2 grid; each wave owns a 32x64 tile = 2x4 fp32 WMMA accumulators.
__global__ void __launch_bounds__(256)
chol_syrk_wmma_kernel(float* M, int off) {
    __shared__ float As[MT * LDP];          // row-panel strip, 34 KB
    __shared__ float Bs[MT * LDP];          // col-panel strip, 34 KB
    int trail = off + NB;
    int tid = threadIdx.x;

    // triangular decode of block id -> (bi >= bj) macro-tile pair
    int b = blockIdx.x;
    int bi = (int)floorf((sqrtf(8.0f * (float)b + 1.0f) - 1.0f) * 0.5f);
    while (((bi + 1) * (bi + 2)) / 2 <= b) ++bi;
    while ((bi * (bi + 1)) / 2 > b) --bi;
    int bj = b - (bi * (bi + 1)) / 2;
    int row0 = trail + bi * MT;
    int col0 = trail + bj * MT;

    // ---- async stage both 128x64 panel strips into LDS (clamp OOB rows) ----
    #pragma unroll
    for (int it = 0; it < 8; ++it) {
        int e  = it * 256 + tid;            // float4 slot in 128x16 grid
        int r  = e >> 4;
        int c4 = (e & 15) << 2;
        int gr = row0 + r; if (gr >= Vc) gr = Vc - 1;
        int gc = col0 + r; if (gc >= Vc) gc = Vc - 1;
        const float* gpA = M + (size_t)gr * Vc + off + c4;
        const float* gpB = M + (size_t)gc * Vc + off + c4;
        unsigned ldsA = (unsigned)(size_t)(void*)&As[r * LDP + c4];
        unsigned ldsB = (unsigned)(size_t)(void*)&Bs[r * LDP + c4];
        asm volatile("global_load_async_to_lds_b128 %0, %1, off"
                     :: "v"(ldsA), "v"(gpA) : "memory");
        asm volatile("global_load_async_to_lds_b128 %0, %1, off"
                     :: "v"(ldsB), "v"(gpB) : "memory");
    }
    asm volatile("s_wait_asynccnt 0x0" ::: "memory");
    __syncthreads();

    // ---- register-blocked WMMA: wave (wr,wc) owns rows wr*32..+31, cols wc*64..+63
    int lane = tid & 31;
    int wid  = tid >> 5;
    int wr   = wid >> 1;                    // 0..3
    int wc   = wid & 1;                     // 0..1
    int m    = lane & 15;
    int hk   = (lane >> 4) << 1;            // K split across lane halves

    v8f acc[2][4];
    #pragma unroll
    for (int r = 0; r < 2; ++r)
        #pragma unroll
        for (int c = 0; c < 4; ++c) acc[r][c] = (v8f){};

    #pragma unroll
    for (int kk = 0; kk < NB; kk += 4) {
        v2f afr[2], bfr[4];
        #pragma unroll
        for (int r = 0; r < 2; ++r) {
            int base = (wr * 32 + r * 16 + m) * LDP + kk + hk;
            afr[r].x = As[base]; afr[r].y = As[base + 1];
        }
        #pragma unroll
        for (int c = 0; c < 4; ++c) {
            int base = (wc * 64 + c * 16 + m) * LDP + kk + hk;
            bfr[c].x = Bs[base]; bfr[c].y = Bs[base + 1];
        }
        #pragma unroll
        for (int r = 0; r < 2; ++r)
            #pragma unroll
            for (int c = 0; c < 4; ++c)
                acc[r][c] = __builtin_amdgcn_wmma_f32_16x16x4_f32(
                    false, afr[r], false, bfr[c], (short)0, acc[r][c], false, false);
    }

    // ---- C -= acc (guard OOB rows/cols; upper-triangle writes are never read)
    int rb = (lane >> 4) << 3;              // C layout: lanes16-31 -> M = t+8
    #pragma unroll
    for (int r = 0; r < 2; ++r) {
        #pragma unroll
        for (int c = 0; c < 4; ++c) {
            int gj = col0 + wc * 64 + c * 16 + m;
            int gi0 = row0 + wr * 32 + r * 16 + rb;
            if (gj < Vc) {
                #pragma unroll
                for (int t = 0; t < 8; ++t) {
                    int gi = gi0 + t;
                    if (gi < Vc) M[(size_t)gi * Vc + gj] -= acc[r][c][t];
                }
            }
        }
    }
}

// ---------------- triangular solves, 4 RHS ----------------
__global__ void fwd_diag_kernel(const float* M, float* bvec, int off) {
    __shared__ float Lkk[NB * NB];
    for (int idx = threadIdx.x; idx < NB * NB; idx += blockDim.x)
        Lkk[idx] = M[(size_t)(off + idx / NB) * Vc + off + (idx % NB)];
    __syncthreads();
    int rhs = threadIdx.x;
    if (rhs < BTOT) {
        float* bp = bvec + (size_t)rhs * Vc + off;
        for (int j = 0; j < NB; ++j) {
            float acc = bp[j];
            for (int t = 0; t < j; ++t) acc -= Lkk[j * NB + t] * bp[t];
            bp[j] = acc / Lkk[j * NB + j];
        }
    }
}

__global__ void fwd_update_kernel(const float* M, float* bvec, int off) {
    int row = off + NB + blockIdx.x * (blockDim.x / BTOT) + threadIdx.x / BTOT;
    int rhs = threadIdx.x % BTOT;
    if (row >= Vc) return;
    const float* Lr = M + (size_t)row * Vc + off;
    const float* y  = bvec + (size_t)rhs * Vc + off;
    float acc = 0.0f;
    #pragma unroll
    for (int t = 0; t < NB; ++t) acc += Lr[t] * y[t];
    bvec[(size_t)rhs * Vc + row] -= acc;
}

__global__ void bwd_diag_kernel(const float* M, float* bvec, int off) {
    __shared__ float Lkk[NB * NB];
    for (int idx = threadIdx.x; idx < NB * NB; idx += blockDim.x)
        Lkk[idx] = M[(size_t)(off + idx / NB) * Vc + off + (idx % NB)];
    __syncthreads();
    int rhs = threadIdx.x;
    if (rhs < BTOT) {
        float* bp = bvec + (size_t)rhs * Vc + off;
        for (int j = NB - 1; j >= 0; --j) {
            float acc = bp[j];
            for (int t = j + 1; t < NB; ++t) acc -= Lkk[t * NB + j] * bp[t];
            bp[j] = acc / Lkk[j * NB + j];
        }
    }
}

__global__ void bwd_update_kernel(const float* M, float* bvec, int off) {
    int i   = blockIdx.x * (blockDim.x / BTOT) + threadIdx.x / BTOT;
    int rhs = threadIdx.x % BTOT;
    if (i >= off) return;
    const float* x = bvec + (size_t)rhs * Vc + off;
    float acc = 0.0f;
    #pragma unroll
    for (int t = 0; t < NB; ++t) acc += M[(size_t)(off + t) * Vc + i] * x[t];
    bvec[(size_t)rhs * Vc + i] -= acc;
}

// out = [ u0 : BTOT*G ][ u : BTOT*G*K ]
__global__ void write_out_kernel(const float* bvec, float* out) {
    int idx = blockIdx.x * blockDim.x + threadIdx.x;
    if (idx >= BTOT * Vc) return;
    int b = idx / Vc, v = idx - b * Vc;
    float val = bvec[(size_t)b * Vc + v];
    out[BTOT * Gc + (size_t)b * Vc + v] = val;          // u
    if ((v % Kc) == 0) out[b * Gc + v / Kc] = val;      // u0
}

// ---------------- host launcher ----------------
extern "C" void kernel_launch(void* const* d_in, const int* in_sizes, int n_in,
                              void* d_out, int out_size, void* d_ws, size_t ws_size,
                              hipStream_t stream) {
    (void)in_sizes; (void)n_in; (void)out_size; (void)ws_size;
    const float* coeffs = (const float*)d_in[0];  // [4,1,1024,5]
    const float* rhs    = (const float*)d_in[1];  // [4,1,1024]
    const float* iv     = (const float*)d_in[2];  // [4,1,32]
    const float* steps0 = (const float*)d_in[3];  // [4,1,31]
    const float* steps1 = (const float*)d_in[4];  // [4,1,31]
    float* out  = (float*)d_out;
    float* M    = (float*)d_ws;                   // Vc*Vc fp32 (100 MB, fits L2)
    float* bvec = M + (size_t)Vc * Vc;            // BTOT*Vc fp32

    size_t nzero = (size_t)Vc * Vc + (size_t)BTOT * Vc;
    zero_ws_kernel<<<4096, 256, 0, stream>>>(M, nzero);
    assemble_AtA_kernel<<<(RTOT + 255) / 256, 256, 0, stream>>>(coeffs, steps0, steps1, M);
    add_alpha_diag_kernel<<<(Vc + 255) / 256, 256, 0, stream>>>(M);
    assemble_rhs_kernel<<<(BTOT * NEQIV + 255) / 256, 256, 0, stream>>>(coeffs, rhs, iv, bvec);

    for (int kb = 0; kb < NBLK; ++kb) {
        int off = kb * NB;
        chol_potf2_kernel<<<1, 256, 0, stream>>>(M, off);
        int rows = Vc - off - NB;
        if (rows > 0) {
            chol_trsm_kernel<<<(rows + 127) / 128, 128, 0, stream>>>(M, off);
            int nmt   = (rows + MT - 1) / MT;
            int tiles = nmt * (nmt + 1) / 2;
            chol_syrk_wmma_kernel<<<tiles, 256, 0, stream>>>(M, off);
        }
    }
    for (int kb = 0; kb < NBLK; ++kb) {
        int off = kb * NB;
        fwd_diag_kernel<<<1, 64, 0, stream>>>(M, bvec, off);
        int rows = Vc - off - NB;
        if (rows > 0)
            fwd_update_kernel<<<(rows * BTOT + 255) / 256, 256, 0, stream>>>(M, bvec, off);
    }
    for (int kb = NBLK - 1; kb >= 0; --kb) {
        int off = kb * NB;
        bwd_diag_kernel<<<1, 64, 0, stream>>>(M, bvec, off);
        if (off > 0)
            bwd_update_kernel<<<(off * BTOT + 255) / 256, 256, 0, stream>>>(M, bvec, off);
    }
    write_out_kernel<<<(BTOT * Vc + 255) / 256, 256, 0, stream>>>(bvec, out);
}